// OptimizedDiverseGAT_51196010169102
// MI455X (gfx1250) — compile-verified
//
#include <hip/hip_runtime.h>
#include <math.h>

// Problem constants (match reference)
#define NN      100000
#define EE      1600000
#define IN_DIM  128
#define HDW     64      // H*D
#define NHD     8       // heads
#define NDD     8       // dim per head
#define NG      64      // graphs
#define NOUT    10
#define NEG_SLOPE 0.2f
#define DIVW    0.1f

typedef __attribute__((ext_vector_type(2))) float v2f;
typedef __attribute__((ext_vector_type(8))) float v8f;

// ---------------------------------------------------------------------------
// fp32 WMMA GEMM: C[nrows,64] = A[nrows,K] @ B[K,64]   (row-major everything)
// One wave32 computes a 16x64 strip via 4 accumulators of V_WMMA_F32_16X16X4_F32.
// ---------------------------------------------------------------------------
__global__ __launch_bounds__(256)
void gemm_wmma_f32(const float* __restrict__ A, const float* __restrict__ B,
                   float* __restrict__ C, int nrows, int K) {
    int wave = (blockIdx.x * blockDim.x + threadIdx.x) >> 5;   // wave-uniform
    int lane = threadIdx.x & 31;
    int ntiles = nrows >> 4;
    if (wave >= ntiles) return;                                // uniform exit
    int row0 = wave << 4;
    int half = lane >> 4;      // 0: K-pair {0,1}, 1: K-pair {2,3}
    int l    = lane & 15;      // M (for A) / N (for B,C)

    v8f c0 = {}, c1 = {}, c2 = {}, c3 = {};
    const float* Arow = A + (size_t)(row0 + l) * K;

    for (int k0 = 0; k0 < K; k0 += 4) {
        int ka = k0 + 2 * half;
        v2f a;
        a.x = Arow[ka];
        a.y = Arow[ka + 1];
        const float* B0 = B + (size_t)ka * HDW;
        const float* B1 = B0 + HDW;
        v2f b;
        b.x = B0[l +  0]; b.y = B1[l +  0];
        c0 = __builtin_amdgcn_wmma_f32_16x16x4_f32(false, a, false, b, (short)0, c0, false, false);
        b.x = B0[l + 16]; b.y = B1[l + 16];
        c1 = __builtin_amdgcn_wmma_f32_16x16x4_f32(false, a, false, b, (short)0, c1, false, false);
        b.x = B0[l + 32]; b.y = B1[l + 32];
        c2 = __builtin_amdgcn_wmma_f32_16x16x4_f32(false, a, false, b, (short)0, c2, false, false);
        b.x = B0[l + 48]; b.y = B1[l + 48];
        c3 = __builtin_amdgcn_wmma_f32_16x16x4_f32(false, a, false, b, (short)0, c3, false, false);
    }
    // C/D layout: VGPR r -> (M = half*8 + r, N = l)
    #pragma unroll
    for (int r = 0; r < 8; ++r) {
        size_t row = (size_t)(row0 + half * 8 + r) * HDW;
        C[row +  0 + l] = c0[r];
        C[row + 16 + l] = c1[r];
        C[row + 32 + l] = c2[r];
        C[row + 48 + l] = c3[r];
    }
}

// ---------------------------------------------------------------------------
// Attention score prep: al[n,h] = <h0[n,h,:], att_src[h,:]>, ar likewise
// ---------------------------------------------------------------------------
__global__ void scores_kernel(const float* __restrict__ h,
                              const float* __restrict__ asrc,
                              const float* __restrict__ adst,
                              float* __restrict__ al, float* __restrict__ ar) {
    int tid = blockIdx.x * blockDim.x + threadIdx.x;
    if (tid >= NN * NHD) return;
    int n = tid >> 3, hh = tid & 7;
    const float* hv = h + (size_t)n * HDW + hh * NDD;
    const float* sa = asrc + hh * NDD;
    const float* sd = adst + hh * NDD;
    float a = 0.f, d = 0.f;
    #pragma unroll
    for (int k = 0; k < NDD; ++k) { a += hv[k] * sa[k]; d += hv[k] * sd[k]; }
    al[tid] = a; ar[tid] = d;
}

// Monotonic float<->uint key for atomic max on floats
__device__ __forceinline__ unsigned fkey(float f) {
    int i = __float_as_int(f);
    return (unsigned)(i ^ ((i >> 31) | 0x80000000));
}
__device__ __forceinline__ float fkey_dec(unsigned k) {
    int i = (k & 0x80000000u) ? (int)(k ^ 0x80000000u) : (int)~k;
    return __int_as_float(i);
}
__device__ __forceinline__ float lrelu(float v) {
    return v > 0.f ? v : NEG_SLOPE * v;
}

// segment-max of leaky_relu scores into key buffer (zero-initialized)
__global__ void edge_max_kernel(const int* __restrict__ ei,
                                const float* __restrict__ al,
                                const float* __restrict__ ar,
                                unsigned* __restrict__ mkey) {
    long long tid = (long long)blockIdx.x * blockDim.x + threadIdx.x;
    if (tid >= (long long)EE * NHD) return;
    int e = (int)(tid >> 3), hh = (int)(tid & 7);
    int src = ei[e], dst = ei[EE + e];
    float s = lrelu(al[src * NHD + hh] + ar[dst * NHD + hh]);
    atomicMax(&mkey[dst * NHD + hh], fkey(s));
}

// decode keys -> float maxima, non-finite (incl. empty segments) -> 0
__global__ void decode_m_kernel(float* __restrict__ mbuf) {
    int tid = blockIdx.x * blockDim.x + threadIdx.x;
    if (tid >= NN * NHD) return;
    float m = fkey_dec(((const unsigned*)mbuf)[tid]);
    if (!(m == m) || fabsf(m) > 3.0e38f) m = 0.f;
    mbuf[tid] = m;
}

// denominator: den[dst,h] += exp(score - m[dst,h])
__global__ void edge_expsum_kernel(const int* __restrict__ ei,
                                   const float* __restrict__ al,
                                   const float* __restrict__ ar,
                                   const float* __restrict__ m,
                                   float* __restrict__ den) {
    long long tid = (long long)blockIdx.x * blockDim.x + threadIdx.x;
    if (tid >= (long long)EE * NHD) return;
    int e = (int)(tid >> 3), hh = (int)(tid & 7);
    int src = ei[e], dst = ei[EE + e];
    int di = dst * NHD + hh;
    float s = lrelu(al[src * NHD + hh] + ar[dst * NHD + hh]);
    atomicAdd(&den[di], expf(s - m[di]));
}

// aggregation: acc[dst,h,:] += softmax_weight * h[src,h,:]
__global__ void edge_agg_kernel(const int* __restrict__ ei,
                                const float* __restrict__ al,
                                const float* __restrict__ ar,
                                const float* __restrict__ m,
                                const float* __restrict__ den,
                                const float* __restrict__ hsrc,
                                float* __restrict__ acc) {
    long long tid = (long long)blockIdx.x * blockDim.x + threadIdx.x;
    if (tid >= (long long)EE * NHD) return;
    int e = (int)(tid >> 3), hh = (int)(tid & 7);
    int src = ei[e], dst = ei[EE + e];
    int di = dst * NHD + hh;
    float s = lrelu(al[src * NHD + hh] + ar[dst * NHD + hh]);
    float a = expf(s - m[di]) / (den[di] + 1e-16f);
    const float* hs = hsrc + (size_t)src * HDW + hh * NDD;
    float* ac = acc + (size_t)dst * HDW + hh * NDD;
    #pragma unroll
    for (int d = 0; d < NDD; ++d) atomicAdd(&ac[d], a * hs[d]);
}

// bias add (in place) + diversity-loss S accumulation + BN sum/sumsq
__global__ __launch_bounds__(256)
void bias_div_stats_kernel(float* __restrict__ hb,
                           const float* __restrict__ bias,
                           const float* __restrict__ P,
                           float* __restrict__ Sout,      // 64 (upper tri used)
                           float* __restrict__ bnsum,
                           float* __restrict__ bnsq) {
    __shared__ float Pl[64], Ss[64], Bs[64], Bq[64];
    int t = threadIdx.x;
    if (t < 64) { Pl[t] = P[t]; Ss[t] = 0.f; Bs[t] = 0.f; Bq[t] = 0.f; }
    __syncthreads();
    int n = blockIdx.x * blockDim.x + t;
    float nh[NHD][NDD];
    if (n < NN) {
        #pragma unroll
        for (int i = 0; i < NHD; ++i) {
            float hv[NDD], pr[NDD];
            #pragma unroll
            for (int d = 0; d < NDD; ++d) {
                size_t idx = (size_t)n * HDW + i * NDD + d;
                float v = hb[idx] + bias[i * NDD + d];
                hb[idx] = v;
                hv[d] = v;
                atomicAdd(&Bs[i * NDD + d], v);
                atomicAdd(&Bq[i * NDD + d], v * v);
            }
            float ss = 0.f;
            #pragma unroll
            for (int k = 0; k < NDD; ++k) {
                float p = 0.f;
                #pragma unroll
                for (int j = 0; j < NDD; ++j) p += hv[j] * Pl[j * NDD + k];
                pr[k] = p; ss += p * p;
            }
            float nrm = fmaxf(sqrtf(ss), 1e-8f);
            #pragma unroll
            for (int k = 0; k < NDD; ++k) nh[i][k] = pr[k] / nrm;
        }
        #pragma unroll
        for (int i = 0; i < NHD; ++i)
            #pragma unroll
            for (int j = i + 1; j < NHD; ++j) {
                float s = 0.f;
                #pragma unroll
                for (int k = 0; k < NDD; ++k) s += nh[i][k] * nh[j][k];
                atomicAdd(&Ss[i * NHD + j], s);
            }
    }
    __syncthreads();
    if (t < 64) {
        atomicAdd(&Sout[t],  Ss[t]);
        atomicAdd(&bnsum[t], Bs[t]);
        atomicAdd(&bnsq[t],  Bq[t]);
    }
}

// BatchNorm (training stats) + ELU
__global__ void bn_elu_kernel(const float* __restrict__ hb,
                              const float* __restrict__ bnsum,
                              const float* __restrict__ bnsq,
                              const float* __restrict__ gamma,
                              const float* __restrict__ beta,
                              float* __restrict__ act) {
    int tid = blockIdx.x * blockDim.x + threadIdx.x;
    if (tid >= NN * HDW) return;
    int c = tid & 63;
    const float invN = 1.0f / (float)NN;
    float mu  = bnsum[c] * invN;
    float var = bnsq[c] * invN - mu * mu;
    float xh = gamma[c] * (hb[tid] - mu) * rsqrtf(var + 1e-5f) + beta[c];
    act[tid] = xh > 0.f ? xh : expm1f(xh);
}

// mean-pool per graph (sum + count via atomics)
__global__ void pool_kernel(const float* __restrict__ act,
                            const int* __restrict__ batch,
                            float* __restrict__ pooled,
                            float* __restrict__ cnt) {
    int tid = blockIdx.x * blockDim.x + threadIdx.x;
    if (tid >= NN * HDW) return;
    int n = tid >> 6, c = tid & 63;
    int g = batch[n];
    atomicAdd(&pooled[g * HDW + c], act[tid]);
    if (c == 0) atomicAdd(&cnt[g], 1.0f);
}

// logits = (pooled / max(cnt,1)) @ fc_W + fc_b
__global__ void logits_kernel(const float* __restrict__ pooled,
                              const float* __restrict__ cnt,
                              const float* __restrict__ fcW,
                              const float* __restrict__ fcb,
                              float* __restrict__ out) {
    int tid = blockIdx.x * blockDim.x + threadIdx.x;
    if (tid >= NG * NOUT) return;
    int g = tid / NOUT, o = tid - g * NOUT;
    float invc = 1.0f / fmaxf(cnt[g], 1.0f);
    float s = fcb[o];
    for (int c = 0; c < HDW; ++c)
        s += pooled[g * HDW + c] * invc * fcW[c * NOUT + o];
    out[tid] = s;
}

// finalize: DIVW * (mean upper-tri (S/N)^2 for each layer)
__global__ void div_final_kernel(const float* __restrict__ Ssum,
                                 float* __restrict__ out) {
    if (threadIdx.x != 0 || blockIdx.x != 0) return;
    float dl = 0.f;
    for (int lyr = 0; lyr < 2; ++lyr) {
        const float* S = Ssum + lyr * 64;
        float a = 0.f;
        for (int i = 0; i < NHD; ++i)
            for (int j = i + 1; j < NHD; ++j) {
                float s = S[i * NHD + j] / (float)NN;
                a += s * s;
            }
        dl += a / 28.0f;   // 28 upper-triangular pairs
    }
    out[NG * NOUT] = DIVW * dl;
}

// ---------------------------------------------------------------------------
extern "C" void kernel_launch(void* const* d_in, const int* in_sizes, int n_in,
                              void* d_out, int out_size, void* d_ws, size_t ws_size,
                              hipStream_t stream) {
    const float* x     = (const float*)d_in[0];
    const int*   ei    = (const int*)  d_in[1];
    const int*   batch = (const int*)  d_in[2];
    const float* W0    = (const float*)d_in[3];
    const float* as0   = (const float*)d_in[4];
    const float* ad0   = (const float*)d_in[5];
    const float* b0    = (const float*)d_in[6];
    const float* g0    = (const float*)d_in[7];
    const float* be0   = (const float*)d_in[8];
    const float* P0    = (const float*)d_in[9];
    const float* W1    = (const float*)d_in[10];
    const float* as1   = (const float*)d_in[11];
    const float* ad1   = (const float*)d_in[12];
    const float* b1    = (const float*)d_in[13];
    const float* g1    = (const float*)d_in[14];
    const float* be1   = (const float*)d_in[15];
    const float* P1    = (const float*)d_in[16];
    const float* fcW   = (const float*)d_in[17];
    const float* fcb   = (const float*)d_in[18];
    float* out = (float*)d_out;

    // workspace layout (floats)
    float* ws     = (float*)d_ws;
    float* h      = ws;                         // N*64  GEMM output
    float* acc    = h    + (size_t)NN * HDW;    // N*64  aggregation / biased feats
    float* act    = acc  + (size_t)NN * HDW;    // N*64  post BN+ELU
    float* al     = act  + (size_t)NN * HDW;    // N*8
    float* ar     = al   + (size_t)NN * NHD;    // N*8
    float* mbuf   = ar   + (size_t)NN * NHD;    // N*8 (keys then floats)
    float* den    = mbuf + (size_t)NN * NHD;    // N*8
    float* pooled = den  + (size_t)NN * NHD;    // G*64
    float* cnt    = pooled + NG * HDW;          // G
    float* Ssum   = cnt  + NG;                  // 2*64
    float* bnsum  = Ssum + 128;                 // 64
    float* bnsq   = bnsum + HDW;                // 64

    const int B = 256;
    const int gemmBlocks  = ((NN / 16) + 7) / 8;           // 8 waves/block
    const int nhBlocks    = (NN * NHD + B - 1) / B;
    const long long ehTot = (long long)EE * NHD;
    const int ehBlocks    = (int)((ehTot + B - 1) / B);
    const int nodeBlocks  = (NN + B - 1) / B;
    const int ncBlocks    = (NN * HDW + B - 1) / B;

    // zero global accumulators (in-stream, graph-capturable)
    hipMemsetAsync(pooled, 0, (size_t)(NG * HDW + NG + 128) * 4, stream);

    // =================== Layer 0 ===================
    hipMemsetAsync(acc,  0, (size_t)NN * HDW * 4, stream);
    hipMemsetAsync(mbuf, 0, (size_t)NN * NHD * 4, stream);
    hipMemsetAsync(den,  0, (size_t)NN * NHD * 4, stream);
    hipMemsetAsync(bnsum, 0, HDW * 4, stream);
    hipMemsetAsync(bnsq,  0, HDW * 4, stream);

    gemm_wmma_f32<<<gemmBlocks, B, 0, stream>>>(x, W0, h, NN, IN_DIM);
    scores_kernel<<<nhBlocks, B, 0, stream>>>(h, as0, ad0, al, ar);
    edge_max_kernel<<<ehBlocks, B, 0, stream>>>(ei, al, ar, (unsigned*)mbuf);
    decode_m_kernel<<<nhBlocks, B, 0, stream>>>(mbuf);
    edge_expsum_kernel<<<ehBlocks, B, 0, stream>>>(ei, al, ar, mbuf, den);
    edge_agg_kernel<<<ehBlocks, B, 0, stream>>>(ei, al, ar, mbuf, den, h, acc);
    bias_div_stats_kernel<<<nodeBlocks, B, 0, stream>>>(acc, b0, P0, Ssum, bnsum, bnsq);
    bn_elu_kernel<<<ncBlocks, B, 0, stream>>>(acc, bnsum, bnsq, g0, be0, act);

    // =================== Layer 1 ===================
    hipMemsetAsync(acc,  0, (size_t)NN * HDW * 4, stream);
    hipMemsetAsync(mbuf, 0, (size_t)NN * NHD * 4, stream);
    hipMemsetAsync(den,  0, (size_t)NN * NHD * 4, stream);
    hipMemsetAsync(bnsum, 0, HDW * 4, stream);
    hipMemsetAsync(bnsq,  0, HDW * 4, stream);

    gemm_wmma_f32<<<gemmBlocks, B, 0, stream>>>(act, W1, h, NN, HDW);
    scores_kernel<<<nhBlocks, B, 0, stream>>>(h, as1, ad1, al, ar);
    edge_max_kernel<<<ehBlocks, B, 0, stream>>>(ei, al, ar, (unsigned*)mbuf);
    decode_m_kernel<<<nhBlocks, B, 0, stream>>>(mbuf);
    edge_expsum_kernel<<<ehBlocks, B, 0, stream>>>(ei, al, ar, mbuf, den);
    edge_agg_kernel<<<ehBlocks, B, 0, stream>>>(ei, al, ar, mbuf, den, h, acc);
    bias_div_stats_kernel<<<nodeBlocks, B, 0, stream>>>(acc, b1, P1, Ssum + 64, bnsum, bnsq);
    bn_elu_kernel<<<ncBlocks, B, 0, stream>>>(acc, bnsum, bnsq, g1, be1, act);

    // =================== Readout ===================
    pool_kernel<<<ncBlocks, B, 0, stream>>>(act, batch, pooled, cnt);
    logits_kernel<<<(NG * NOUT + B - 1) / B, B, 0, stream>>>(pooled, cnt, fcW, fcb, out);
    div_final_kernel<<<1, 32, 0, stream>>>(Ssum, out);

    (void)in_sizes; (void)n_in; (void)out_size; (void)ws_size;
}